// AttnDecoderRNN_62706522522012
// MI455X (gfx1250) — compile-verified
//
#include <hip/hip_runtime.h>
#include <math.h>

typedef __attribute__((ext_vector_type(2))) float v2f;
typedef __attribute__((ext_vector_type(4))) float v4f;
typedef __attribute__((ext_vector_type(8))) float v8f;

#define HD 1024
#define VO 50257
#define LN 250

// ---------------------------------------------------------------------------
// Kernel 1: embedding + attention scores + softmax + attn_applied + combined
// Single block, 256 threads (8 waves, wave32).
// ---------------------------------------------------------------------------
__global__ __launch_bounds__(256) void attn_fuse_kernel(
    const int* __restrict__ input, const float* __restrict__ hidden,
    const float* __restrict__ emb, const float* __restrict__ attn_W,
    const float* __restrict__ attn_b, const float* __restrict__ enc,
    float* __restrict__ attn_out, float* __restrict__ combined)
{
    __shared__ float s_sc[256];
    __shared__ float s_red[256];
    __shared__ float s_aw[256];
    const int tid  = threadIdx.x;
    const int lane = tid & 31;
    const int wave = tid >> 5;
    const float* embedded = emb + (size_t)input[0] * HD;
    const float* h0 = hidden;   // (1,1,H) flat

    // attention scores: one wave per row, coalesced along K, butterfly reduce
    for (int row = wave; row < LN; row += 8) {
        const float* wr = attn_W + (size_t)row * (2 * HD);
        float acc = 0.f;
        for (int k = lane; k < 2 * HD; k += 32) {
            float xv = (k < HD) ? embedded[k] : h0[k - HD];
            acc += xv * wr[k];
        }
        for (int off = 16; off > 0; off >>= 1)
            acc += __shfl_xor(acc, off, 32);
        if (lane == 0) s_sc[row] = acc + attn_b[row];
    }
    __syncthreads();

    // block softmax over 250 scores
    float v = (tid < LN) ? s_sc[tid] : -3.4e38f;
    s_red[tid] = v; __syncthreads();
    for (int s = 128; s > 0; s >>= 1) {
        if (tid < s) s_red[tid] = fmaxf(s_red[tid], s_red[tid + s]);
        __syncthreads();
    }
    float mx = s_red[0]; __syncthreads();
    float e = (tid < LN) ? __expf(v - mx) : 0.f;
    s_red[tid] = e; __syncthreads();
    for (int s = 128; s > 0; s >>= 1) {
        if (tid < s) s_red[tid] += s_red[tid + s];
        __syncthreads();
    }
    float aw = e * (1.f / s_red[0]);
    if (tid < LN) { s_aw[tid] = aw; attn_out[tid] = aw; }
    __syncthreads();

    // attn_applied = aw @ enc (coalesced across tid) ; combined = [embedded | applied]
    for (int j = tid; j < HD; j += 256) {
        float acc = 0.f;
        for (int l = 0; l < LN; ++l)
            acc += s_aw[l] * enc[(size_t)l * HD + j];
        combined[HD + j] = acc;
        combined[j] = embedded[j];
    }
}

// ---------------------------------------------------------------------------
// Generic GEMV via v_wmma_f32_16x16x4_f32:  y[m] = sum_k W0[m,k]*v0[k]
//                                          (+ sum_k W1[m,k]*v1[k]) + biases
// 256 threads = 8 waves; each wave owns a 16-row tile; vector staged in LDS.
//
// K-permutation trick: dot products are K-order independent, so we remap
// memory indices to hardware K slots such that each lane's A fetch is a
// single contiguous 16B global_load_b128 feeding TWO wmmas:
//   lanes 0-15  (K slots 0,1) own mem floats [k   , k+4)
//   lanes 16-31 (K slots 2,3) own mem floats [k+4 , k+8)
//   wmma #0 uses elements {0,1}, wmma #1 uses elements {2,3}.
// B (the vector, from LDS) uses the identical mapping, so the result is exact.
// Weight loads are non-temporal: streamed once, out_W (206MB) > L2 (192MB).
// ---------------------------------------------------------------------------
__device__ __forceinline__ void wmma_k8(v8f& c, const float* __restrict__ wrow,
                                        const float* __restrict__ svec,
                                        int k, int kb4)
{
    v4f aw4 = __builtin_nontemporal_load((const v4f*)(wrow + k + kb4));
    v4f bw4 = *(const v4f*)(svec + k + kb4);
    v2f a0 = __builtin_shufflevector(aw4, aw4, 0, 1);
    v2f a1 = __builtin_shufflevector(aw4, aw4, 2, 3);
    v2f b0 = __builtin_shufflevector(bw4, bw4, 0, 1);
    v2f b1 = __builtin_shufflevector(bw4, bw4, 2, 3);
    c = __builtin_amdgcn_wmma_f32_16x16x4_f32(false, a0, false, b0,
                                              (short)0, c, false, false);
    c = __builtin_amdgcn_wmma_f32_16x16x4_f32(false, a1, false, b1,
                                              (short)0, c, false, false);
}

__global__ __launch_bounds__(256) void wmma_gemv_kernel(
    const float* __restrict__ W0, const float* __restrict__ v0, int K0,
    const float* __restrict__ W1, const float* __restrict__ v1, int K1,
    const float* __restrict__ b0, const float* __restrict__ b1,
    float* __restrict__ y, int M, int relu_act)
{
    __shared__ float sv[2 * HD];
    const int tid = threadIdx.x;
    const int Ktot = K0 + K1;
    for (int i = tid; i < Ktot; i += 256)
        sv[i] = (i < K0) ? v0[i] : v1[i - K0];
    __syncthreads();

    const int lane = tid & 31;
    const int wave = tid >> 5;
    const int row0 = (blockIdx.x * 8 + wave) * 16;
    if (row0 >= M) return;                    // wave-uniform skip

    const int mloc = lane & 15;
    const int kb4  = (lane >> 4) * 4;         // mem sub-block: [0,4) vs [4,8)
    const int rowm = min(row0 + mloc, M - 1); // clamp tail rows (loads only)

    v8f c = {0.f, 0.f, 0.f, 0.f, 0.f, 0.f, 0.f, 0.f};
    {
        const float* wr = W0 + (size_t)rowm * K0;
        for (int k = 0; k < K0; k += 8)
            wmma_k8(c, wr, sv, k, kb4);
    }
    if (W1 != nullptr) {
        const float* wr = W1 + (size_t)rowm * K1;
        for (int k = 0; k < K1; k += 8)
            wmma_k8(c, wr, sv + K0, k, kb4);
    }

    if (mloc == 0) {
        const int rbase = row0 + (lane >> 4) * 8;  // lane0 -> M 0..7, lane16 -> M 8..15
        for (int r = 0; r < 8; ++r) {
            int row = rbase + r;
            if (row < M) {
                float val = c[r];
                if (b0) val += b0[row];
                if (b1) val += b1[row];
                if (relu_act) val = fmaxf(val, 0.f);
                y[row] = val;
            }
        }
    }
}

// ---------------------------------------------------------------------------
// Kernel 4: LSTM cell elementwise (torch gate order i,f,g,o)
// ---------------------------------------------------------------------------
__global__ __launch_bounds__(256) void lstm_kernel(
    const float* __restrict__ gates, const float* __restrict__ cell,
    float* __restrict__ h_out, float* __restrict__ c_out,
    float* __restrict__ h_ws)
{
    int j = blockIdx.x * 256 + threadIdx.x;
    if (j >= HD) return;
    float ig = gates[j];
    float fg = gates[HD + j];
    float gg = gates[2 * HD + j];
    float og = gates[3 * HD + j];
    ig = 1.f / (1.f + __expf(-ig));
    fg = 1.f / (1.f + __expf(-fg));
    gg = tanhf(gg);
    og = 1.f / (1.f + __expf(-og));
    float cn = fg * cell[j] + ig * gg;
    float hn = og * tanhf(cn);
    h_out[j] = hn;
    c_out[j] = cn;
    h_ws[j]  = hn;
}

extern "C" void kernel_launch(void* const* d_in, const int* in_sizes, int n_in,
                              void* d_out, int out_size, void* d_ws, size_t ws_size,
                              hipStream_t stream)
{
    const int*   input  = (const int*)  d_in[0];
    const float* hidden = (const float*)d_in[1];
    const float* cell   = (const float*)d_in[2];
    const float* enc    = (const float*)d_in[3];
    const float* emb    = (const float*)d_in[4];
    const float* attn_W = (const float*)d_in[5];
    const float* attn_b = (const float*)d_in[6];
    const float* comb_W = (const float*)d_in[7];
    const float* comb_b = (const float*)d_in[8];
    const float* W_ih   = (const float*)d_in[9];
    const float* b_ih   = (const float*)d_in[10];
    const float* W_hh   = (const float*)d_in[11];
    const float* b_hh   = (const float*)d_in[12];
    const float* out_W  = (const float*)d_in[13];
    const float* out_b  = (const float*)d_in[14];

    float* out    = (float*)d_out;
    float* logits = out;                 // [0, V)
    float* h_out  = out + VO;            // [V, V+H)
    float* c_out  = out + VO + HD;       // [V+H, V+2H)
    float* aw_out = out + VO + 2 * HD;   // [V+2H, V+2H+L)

    float* ws       = (float*)d_ws;
    float* combined = ws;                // 2H
    float* x_ws     = ws + 2 * HD;       // H
    float* gates_ws = ws + 3 * HD;       // 4H
    float* h_ws     = ws + 7 * HD;       // H

    // 1) attention + combined (single block; serial dependency head)
    attn_fuse_kernel<<<1, 256, 0, stream>>>(input, hidden, emb, attn_W, attn_b,
                                            enc, aw_out, combined);
    // 2) x = relu(comb_W @ combined + comb_b)      M=1024, K=2048
    wmma_gemv_kernel<<<(HD + 127) / 128, 256, 0, stream>>>(
        comb_W, combined, 2 * HD, nullptr, nullptr, 0,
        comb_b, nullptr, x_ws, HD, 1);
    // 3) gates = W_ih@x + W_hh@h0 + b_ih + b_hh    M=4096, K=1024(x2)
    wmma_gemv_kernel<<<(4 * HD + 127) / 128, 256, 0, stream>>>(
        W_ih, x_ws, HD, W_hh, hidden, HD,
        b_ih, b_hh, gates_ws, 4 * HD, 0);
    // 4) LSTM elementwise -> h_new, c_new
    lstm_kernel<<<(HD + 255) / 256, 256, 0, stream>>>(
        gates_ws, cell, h_out, c_out, h_ws);
    // 5) logits = out_W @ h_new + out_b            M=50257, K=1024 (206 MB stream)
    wmma_gemv_kernel<<<(VO + 127) / 128, 256, 0, stream>>>(
        out_W, h_ws, HD, nullptr, nullptr, 0,
        out_b, nullptr, logits, VO, 0);
}